// SE_loss_w_threshold_60773787239040
// MI455X (gfx1250) — compile-verified
//
#include <hip/hip_runtime.h>
#include <hip/hip_bf16.h>

#define NUSER 8
#define BATCH 16384
#define NNT   128
#define HALF_NT 64
#define T_THRESH 0.3f
#define LOG2_10  3.3219280948873623f

typedef float v2f __attribute__((ext_vector_type(2)));
typedef float v8f __attribute__((ext_vector_type(8)));

// gacc layout (in d_ws): [0..7] = sum over batch of R_user per user, [8] = sum over batch of Rp
__global__ void se_zero_kernel(float* gacc) {
    if (threadIdx.x < 9) gacc[threadIdx.x] = 0.0f;
}

// One wave handles a pair of batch samples per iteration; 16x16 WMMA tile holds
// both samples' 8x8 complex Gram matrices (as Re / Im fp32 GEMMs) on the diagonal blocks.
__global__ __launch_bounds__(256) void se_main_kernel(const float* __restrict__ h,
                                                      const float* __restrict__ v,
                                                      const float* __restrict__ Np,
                                                      float* __restrict__ gacc) {
    __shared__ float iallS[8][2][8][8];  // [wave][sample][i][j] = |A[i,j]|^2
    __shared__ float vssS[8][32];        // per-lane v sum-of-squares
    __shared__ float blkacc[9];          // block-level accumulators

    const int tid  = threadIdx.x;
    if (tid < 9) blkacc[tid] = 0.0f;
    __syncthreads();

    const int wave = tid >> 5;
    const int lane = tid & 31;
    const int half = lane >> 4;      // K sub-group: 0 -> K{0,1}, 1 -> K{2,3}
    const int m    = lane & 15;      // A row / B column within the 16x16 tile
    const int user = m & 7;
    const int ssel = m >> 3;         // 0 -> first sample of pair, 1 -> second
    const int koff = half * 2;
    const float Nval = Np[0];

    const int totalWaves = (int)gridDim.x * 8;
    const int npairs = BATCH / 2;

    for (int p = blockIdx.x * 8 + wave; p < npairs; p += totalWaves) {
        const int b = 2 * p + ssel;
        const long base = ((long)user * BATCH + b) * NNT;

        v8f accRe = {};  // Re(A) = h_lo.v_lo + h_hi.v_hi
        v8f accP1 = {};  // h_hi.v_lo
        v8f accP2 = {};  // h_lo.v_hi   (Im = P1 - P2)
        float vss = 0.0f;

#pragma unroll
        for (int kb = 0; kb < 16; ++kb) {
            const int t = 4 * kb + koff;
            v2f alo = *(const v2f*)(h + base + t);
            v2f ahi = *(const v2f*)(h + base + HALF_NT + t);
            v2f blo = *(const v2f*)(v + base + t);
            v2f bhi = *(const v2f*)(v + base + HALF_NT + t);
            vss += blo.x * blo.x + blo.y * blo.y + bhi.x * bhi.x + bhi.y * bhi.y;
            accRe = __builtin_amdgcn_wmma_f32_16x16x4_f32(false, alo, false, blo,
                                                          (short)0, accRe, false, false);
            accRe = __builtin_amdgcn_wmma_f32_16x16x4_f32(false, ahi, false, bhi,
                                                          (short)0, accRe, false, false);
            accP1 = __builtin_amdgcn_wmma_f32_16x16x4_f32(false, ahi, false, blo,
                                                          (short)0, accP1, false, false);
            accP2 = __builtin_amdgcn_wmma_f32_16x16x4_f32(false, alo, false, bhi,
                                                          (short)0, accP2, false, false);
        }

        // |A|^2 per (row r, col lane): C layout -> VGPR r: lanes0-15 M=r, lanes16-31 M=r+8
        float iq[8];
#pragma unroll
        for (int r = 0; r < 8; ++r) {
            float im = accP1[r] - accP2[r];
            float re = accRe[r];
            iq[r] = re * re + im * im;
        }

        vssS[wave][lane] = vss;
        if (lane < 8) {
            // sample 0 columns j=lane, rows 0..7 (regs 0..7)
#pragma unroll
            for (int r = 0; r < 8; ++r) iallS[wave][0][r][lane] = iq[r];
        } else if (lane >= 24) {
            // sample 1 columns j=lane-24 (N=lane-16 in 8..15), rows 8..15 (regs 0..7)
            const int j = lane - 24;
#pragma unroll
            for (int r = 0; r < 8; ++r) iallS[wave][1][r][j] = iq[r];
        }
        __syncthreads();   // uniform trip count across block -> legal

        if (lane < 16) {
            const int s = lane >> 3;
            const int i = lane & 7;
            float rowsum = 0.0f;
#pragma unroll
            for (int j = 0; j < 8; ++j) rowsum += iallS[wave][s][i][j];
            const float diag = iallS[wave][s][i][i];
            float n2 = 0.0f;
#pragma unroll
            for (int n = 0; n < 8; ++n) {
                const int c = s * 8 + n;
                n2 += vssS[wave][c] + vssS[wave][c + 16];
            }
            const float S    = (8.0f / n2) * diag;        // scale^2 * |A_ii|^2
            const float Ip   = rowsum - diag;             // unscaled interference
            const float sinr = S / (Ip + Nval);
            const float R    = __log2f(1.0f + sinr);
            const float rp   = exp2f((T_THRESH - R) * LOG2_10) - R;
            atomicAdd(&blkacc[i], R);
            atomicAdd(&blkacc[8], rp);
        }
        __syncthreads();
    }

    __syncthreads();
    if (tid < 9) atomicAdd(&gacc[tid], blkacc[tid]);
}

__global__ void se_final_kernel(const float* __restrict__ gacc, float* __restrict__ out) {
    if (threadIdx.x == 0) {
        const float inv = 1.0f / (float)BATCH;
        float sumR = 0.0f;
#pragma unroll
        for (int i = 0; i < 8; ++i) {
            float r = gacc[i] * inv;
            out[1 + i] = r;
            sumR += r;
        }
        out[0] = gacc[8] * inv;  // mean(Rp)
        out[9] = sumR;           // sum of per-user means
    }
}

extern "C" void kernel_launch(void* const* d_in, const int* in_sizes, int n_in,
                              void* d_out, int out_size, void* d_ws, size_t ws_size,
                              hipStream_t stream) {
    const float* h  = (const float*)d_in[0];
    const float* v  = (const float*)d_in[1];
    const float* Np = (const float*)d_in[2];
    float* out  = (float*)d_out;
    float* gacc = (float*)d_ws;

    se_zero_kernel<<<1, 32, 0, stream>>>(gacc);
    se_main_kernel<<<64, 256, 0, stream>>>(h, v, Np, gacc);
    se_final_kernel<<<1, 32, 0, stream>>>(gacc, out);
}